// GNNEncoder_26431228739921
// MI455X (gfx1250) — compile-verified
//
#include <hip/hip_runtime.h>

// ---------------------------------------------------------------------------
// GraphConv x2 for MI455X (gfx1250, wave32).
//
//   layer(x) = scatter_add(w_e * x[src] -> dst) @ W_rel + b + x @ W_root
// Rewritten (linear map commutes with segment-sum):
//   y = x @ W_rel            (WMMA f32 16x16x4 dense GEMM)
//   z = x @ W_root + b       (same pass, second accumulator set)
//   out = z + scatter_add(w_e * y[src] -> dst)   (L2-resident f32 atomics)
// ReLU between layers is fused into the layer-2 GEMM's input load.
// ---------------------------------------------------------------------------

typedef __attribute__((ext_vector_type(2))) float v2f;
typedef __attribute__((ext_vector_type(8))) float v8f;

#define FDIM 64

// ---------------------------------------------------------------------------
// Dual GEMM: out_rel = act(in) @ Wrel ; out_root = act(in) @ Wroot + bias
// in:[nrows,64] row-major, W:[64,64] row-major (k-major), one wave per
// 16-row tile, 8 waves per block. Uses V_WMMA_F32_16X16X4_F32.
//
// Fragment layouts (ISA 7.12.2, f32):
//   A 16x4:  lanes 0-15 row M=l hold K={k,k+1} in (v0,v1); lanes 16-31 K={k+2,k+3}
//   B 4x16:  lanes 0-15 col N=l hold K={k,k+1}; lanes 16-31 K={k+2,k+3}
//   C 16x16: VGPR r -> (M=r, N=lane) lanes 0-15 ; (M=r+8, N=lane-16) lanes 16-31
// ---------------------------------------------------------------------------
template <bool RELU>
__global__ __launch_bounds__(256) void gemm_dual_wmma(
    const float* __restrict__ in,
    const float* __restrict__ Wrel,
    const float* __restrict__ Wroot,
    const float* __restrict__ bias,
    float* __restrict__ out_rel,
    float* __restrict__ out_root,
    int nrows)
{
    const int lane = threadIdx.x & 31;
    const int wave = threadIdx.x >> 5;
    const int tile = blockIdx.x * 8 + wave;   // 16-row tile handled by this wave
    const int row0 = tile * 16;
    if (row0 >= nrows) return;                // wave-uniform: EXEC stays all-1s

    const int half = lane >> 4;               // 0: lanes 0-15, 1: lanes 16-31
    const int l15  = lane & 15;

    const v8f zero = {0.f, 0.f, 0.f, 0.f, 0.f, 0.f, 0.f, 0.f};
    v8f accR[4] = {zero, zero, zero, zero};   // 4 col-tiles vs W_rel
    v8f accO[4] = {zero, zero, zero, zero};   // 4 col-tiles vs W_root

    const float* arow = in + (size_t)(row0 + l15) * FDIM;

    for (int kt = 0; kt < 16; ++kt) {         // K = 64 in steps of 4
        const int k = kt * 4 + half * 2;
        float a0 = arow[k];
        float a1 = arow[k + 1];
        if (RELU) {
            a0 = fmaxf(a0, 0.0f);
            a1 = fmaxf(a1, 0.0f);
        }
        v2f a;
        a.x = a0;
        a.y = a1;

#pragma unroll
        for (int nt = 0; nt < 4; ++nt) {
            const int col = nt * 16 + l15;
            v2f br, bo;
            br.x = Wrel[(size_t)k * FDIM + col];
            br.y = Wrel[(size_t)(k + 1) * FDIM + col];
            bo.x = Wroot[(size_t)k * FDIM + col];
            bo.y = Wroot[(size_t)(k + 1) * FDIM + col];

            accR[nt] = __builtin_amdgcn_wmma_f32_16x16x4_f32(
                false, a, false, br, (short)0, accR[nt], false, false);
            accO[nt] = __builtin_amdgcn_wmma_f32_16x16x4_f32(
                false, a, false, bo, (short)0, accO[nt], false, false);
        }
    }

#pragma unroll
    for (int nt = 0; nt < 4; ++nt) {
        const int col = nt * 16 + l15;
        const float bval = bias[col];
#pragma unroll
        for (int r = 0; r < 8; ++r) {
            const int row = row0 + half * 8 + r;
            out_rel[(size_t)row * FDIM + col]  = accR[nt][r];
            out_root[(size_t)row * FDIM + col] = accO[nt][r] + bval;
        }
    }
}

// ---------------------------------------------------------------------------
// Edge scatter: acc[dst,:] += w_e * y[src,:]. One thread covers 4 features
// (float4 gather + 4 agent-scope f32 atomics -> global_atomic_add_f32).
// y (25.6MB) and acc (25.6MB) both fit in the 192MB L2, so the random
// gathers/atomics resolve in L2, not HBM.
// ---------------------------------------------------------------------------
__global__ __launch_bounds__(256) void edge_scatter(
    const int* __restrict__ ei,      // [2,E] flat: src = ei[e], dst = ei[E+e]
    const float* __restrict__ ew,    // [E]
    const float* __restrict__ y,     // [N,64]
    float* __restrict__ acc,         // [N,64], pre-filled with root term
    int E)
{
    const long long total  = (long long)E * 16;   // 16 thread-slots per edge
    const long long stride = (long long)gridDim.x * blockDim.x;
    for (long long idx = (long long)blockIdx.x * blockDim.x + threadIdx.x;
         idx < total; idx += stride) {
        const int e  = (int)(idx >> 4);
        const int f4 = ((int)idx & 15) << 2;

        const int   src = ei[e];
        const int   dst = ei[E + e];
        const float w   = ew[e];

        const float4 v = *(const float4*)(y + (size_t)src * FDIM + f4);
        float* p = acc + (size_t)dst * FDIM + f4;
        __hip_atomic_fetch_add(p + 0, w * v.x, __ATOMIC_RELAXED, __HIP_MEMORY_SCOPE_AGENT);
        __hip_atomic_fetch_add(p + 1, w * v.y, __ATOMIC_RELAXED, __HIP_MEMORY_SCOPE_AGENT);
        __hip_atomic_fetch_add(p + 2, w * v.z, __ATOMIC_RELAXED, __HIP_MEMORY_SCOPE_AGENT);
        __hip_atomic_fetch_add(p + 3, w * v.w, __ATOMIC_RELAXED, __HIP_MEMORY_SCOPE_AGENT);
    }
}

extern "C" void kernel_launch(void* const* d_in, const int* in_sizes, int n_in,
                              void* d_out, int out_size, void* d_ws, size_t ws_size,
                              hipStream_t stream)
{
    // setup_inputs order: x, edge_index, edge_weight, W1_rel, b1, W1_root,
    //                     W2_rel, b2, W2_root
    const float* x   = (const float*)d_in[0];
    const int*   ei  = (const int*)d_in[1];
    const float* ew  = (const float*)d_in[2];
    const float* W1r = (const float*)d_in[3];
    const float* b1  = (const float*)d_in[4];
    const float* W1o = (const float*)d_in[5];
    const float* W2r = (const float*)d_in[6];
    const float* b2  = (const float*)d_in[7];
    const float* W2o = (const float*)d_in[8];

    const int N = in_sizes[0] / FDIM;   // 100000
    const int E = in_sizes[2];          // 1600000

    float* y   = (float*)d_ws;                  // [N,64] rel-transformed feats
    float* h   = y + (size_t)N * FDIM;          // [N,64] layer accumulator
    float* out = (float*)d_out;                 // [N,64] final output

    const int tiles   = (N + 15) / 16;
    const int gblocks = (tiles + 7) / 8;        // 8 waves (16-row tiles) / block

    const long long sthreads = (long long)E * 16;
    int sblocks = (int)((sthreads + 255) / 256);
    if (sblocks > 16384) sblocks = 16384;       // grid-stride covers the rest

    // Layer 1: y = x@W1_rel ; h = x@W1_root + b1 ; h += scatter(w * y[src])
    gemm_dual_wmma<false><<<gblocks, 256, 0, stream>>>(x, W1r, W1o, b1, y, h, N);
    edge_scatter<<<sblocks, 256, 0, stream>>>(ei, ew, y, h, E);

    // Layer 2 (ReLU fused into GEMM input load):
    // y = relu(h)@W2_rel ; out = relu(h)@W2_root + b2 ; out += scatter(w * y[src])
    gemm_dual_wmma<true><<<gblocks, 256, 0, stream>>>(h, W2r, W2o, b2, y, out, N);
    edge_scatter<<<sblocks, 256, 0, stream>>>(ei, ew, y, out, E);
}